// TextMoCo_19198503813796
// MI455X (gfx1250) — compile-verified
//
#include <hip/hip_runtime.h>
#include <hip/hip_bf16.h>
#include <math.h>

typedef __attribute__((ext_vector_type(2))) float v2f;
typedef __attribute__((ext_vector_type(8))) float v8f;

#define NCLASS 10
#define NEG_INF_F (-1e9f)
#define INV_T (1.0f / 0.07f)

// ---------------- workspace layout (bytes) ----------------
#define OFF_SUMS  0            // 2*10*128 floats = 10240 B
#define OFF_CNT   16384        // 20 ints
#define OFF_CODE  32768        // K bytes (131072)
#define OFF_MAT   (OFF_CODE + 131072)          // 128*32 floats = 16384 B
#define OFF_QN    (OFF_MAT + 32768)            // 1024*128 floats = 512 KB
#define OFF_SIM   (OFF_QN + 1024*128*4)        // 1024*32 floats = 128 KB

// ---------------- kernel 0: zero accumulators ----------------
__global__ void zero_kernel(float* sums, int* counts, int nsum) {
    int i = threadIdx.x + blockIdx.x * blockDim.x;
    for (int j = i; j < nsum; j += blockDim.x * gridDim.x) sums[j] = 0.0f;
    if (i < 2 * NCLASS) counts[i] = 0;
}

// ---------------- kernel 1: per-k code byte + mask counts ----------------
// code[k] = (queue_pred[k] << 1) | (queue_pred[k]==queue_label[k])
__global__ void prep_kernel(const int* __restrict__ qp, const int* __restrict__ ql,
                            unsigned char* __restrict__ code, int* __restrict__ counts, int K) {
    __shared__ int lc[2 * NCLASS];
    if (threadIdx.x < 2 * NCLASS) lc[threadIdx.x] = 0;
    __syncthreads();
    int k = blockIdx.x * blockDim.x + threadIdx.x;
    if (k < K) {
        int p = qp[k];
        int corr = (p == ql[k]) ? 1 : 0;
        code[k] = (unsigned char)((p << 1) | corr);
        atomicAdd(&lc[corr ? p : (NCLASS + p)], 1);   // [0..9]=pos counts, [10..19]=neg counts
    }
    __syncthreads();
    if (threadIdx.x < 2 * NCLASS && lc[threadIdx.x] != 0)
        atomicAdd(&counts[threadIdx.x], lc[threadIdx.x]);
}

// ---------------- kernel 2: masked per-class column sums ----------------
// grid: (D, KCHUNKS, 2). block: 256 threads, coalesced along k.
// sums[(sel*10 + c)*D + d] += sum over selected k of queue[d*K + k]
__global__ void colsum_kernel(const float* __restrict__ pos_queue,
                              const float* __restrict__ neg_queue,
                              const unsigned char* __restrict__ code,
                              float* __restrict__ sums, int K, int D) {
    const int d     = blockIdx.x;
    const int chunk = blockIdx.y;
    const int sel   = blockIdx.z;            // 0 = pos (want correct), 1 = neg (want !correct)
    const float* src = (sel == 0 ? pos_queue : neg_queue) + (size_t)d * (size_t)K;
    const int kPer = K / gridDim.y;
    const int k0 = chunk * kPer;
    const int want = (sel == 0) ? 1 : 0;

    float acc[NCLASS];
#pragma unroll
    for (int c = 0; c < NCLASS; ++c) acc[c] = 0.0f;

    for (int k = k0 + threadIdx.x; k < k0 + kPer; k += blockDim.x) {
        unsigned char cd = code[k];
        int cls  = cd >> 1;
        int corr = cd & 1;
        float v = src[k];
        float g = (corr == want) ? v : 0.0f;
#pragma unroll
        for (int c = 0; c < NCLASS; ++c) acc[c] += (cls == c) ? g : 0.0f;
    }

    __shared__ float red[NCLASS][256];
    const int tid = threadIdx.x;
#pragma unroll
    for (int c = 0; c < NCLASS; ++c) red[c][tid] = acc[c];
    __syncthreads();
    for (int s = 128; s > 0; s >>= 1) {
        if (tid < s) {
#pragma unroll
            for (int c = 0; c < NCLASS; ++c) red[c][tid] += red[c][tid + s];
        }
        __syncthreads();
    }
    if (tid < NCLASS) atomicAdd(&sums[(sel * NCLASS + tid) * D + d], red[tid][0]);
}

// ---------------- kernel 3: row L2-normalize q ----------------
__global__ void qnorm_kernel(const float* __restrict__ q, float* __restrict__ qn, int D) {
    const int b = blockIdx.x;
    const int t = threadIdx.x;               // blockDim == D == 128
    __shared__ float sm[128];
    float v = q[(size_t)b * D + t];
    sm[t] = v * v;
    __syncthreads();
    for (int s = 64; s > 0; s >>= 1) {
        if (t < s) sm[t] += sm[t + s];
        __syncthreads();
    }
    float nrm = fmaxf(sqrtf(sm[0]), 1e-12f);
    qn[(size_t)b * D + t] = v / nrm;
}

// ---------------- kernel 4: build Mat[128][32] ----------------
// cols 0..9 = pos class sums, 16..25 = neg class sums, rest zero
__global__ void matbuild_kernel(const float* __restrict__ sums, float* __restrict__ Mat, int D) {
    const int d = blockIdx.x;                // 0..D-1
    const int j = threadIdx.x;               // 0..31
    const int jj = j & 15;
    const int sel = j >> 4;
    float v = 0.0f;
    if (jj < NCLASS) v = sums[(sel * NCLASS + jj) * D + d];
    Mat[d * 32 + j] = v;
}

// ---------------- kernel 5: WMMA GEMM qn[1024,128] x Mat[128,32] ----------------
// wave-per-tile: 64 row tiles x 2 col halves = 128 waves; K loop = 32 steps of k=4.
__global__ void wmma_gemm_kernel(const float* __restrict__ qn, const float* __restrict__ Mat,
                                 float* __restrict__ simout) {
    constexpr int D = 128;
    const int lane  = threadIdx.x & 31;
    const int wib   = threadIdx.x >> 5;
    const int w     = blockIdx.x * (blockDim.x >> 5) + wib;
    const int rt    = w >> 1;                // row tile 0..63
    const int ch    = w & 1;                 // column half 0..1
    const int m     = lane & 15;             // M (A) / N (B,C,D)
    const int koff  = (lane >> 4) * 2;       // K sub-offset per 32-bit A/B layout
    const int mrow  = rt * 16 + m;
    const int ncol  = ch * 16 + m;

    v8f acc = {};
#pragma unroll
    for (int s = 0; s < D / 4; ++s) {
        const int kb = s * 4 + koff;
        v2f a = *(const v2f*)(qn + (size_t)mrow * D + kb);   // A[m][kb], A[m][kb+1]
        v2f b;
        b.x = Mat[(kb + 0) * 32 + ncol];                     // B[kb][n]
        b.y = Mat[(kb + 1) * 32 + ncol];                     // B[kb+1][n]
        acc = __builtin_amdgcn_wmma_f32_16x16x4_f32(
            /*neg_a=*/false, a, /*neg_b=*/false, b,
            /*c_mod=*/(short)0, acc, /*reuse_a=*/false, /*reuse_b=*/false);
    }
    const int rbase = rt * 16 + 8 * (lane >> 4);
#pragma unroll
    for (int v = 0; v < 8; ++v)
        simout[(size_t)(rbase + v) * 32 + ncol] = acc[v];
}

// ---------------- kernel 6: finalize loss + accuracy ----------------
__global__ void finalize_kernel(const float* __restrict__ simout, const int* __restrict__ counts,
                                const int* __restrict__ predictions, float* __restrict__ out, int B) {
    const int b = threadIdx.x;               // blockDim == 1024 == B
    __shared__ float red[1024];
    float good = 0.0f;
    if (b < B) {
        int c  = predictions[b];
        float sp = simout[(size_t)b * 32 + c];
        float sn = simout[(size_t)b * 32 + 16 + c];
        int cp = counts[c];
        int cn = counts[NCLASS + c];
        float lp = (cp > 0) ? sp / (float)max(cp, 1) : NEG_INF_F;
        float ln_ = (cn > 0) ? sn / (float)max(cn, 1) : NEG_INF_F;
        lp *= INV_T;
        ln_ *= INV_T;
        float mx  = fmaxf(lp, ln_);
        float lse = mx + logf(expf(lp - mx) + expf(ln_ - mx));
        out[b] = lse - lp;                   // -log_softmax(logits)[:,0]
        good = (ln_ > lp) ? 0.0f : 1.0f;     // argmax==0 (ties -> 0)
    }
    red[b] = good;
    __syncthreads();
    for (int s = 512; s > 0; s >>= 1) {
        if (b < s) red[b] += red[b + s];
        __syncthreads();
    }
    if (b == 0) out[B] = red[0] / (float)B;
}

extern "C" void kernel_launch(void* const* d_in, const int* in_sizes, int n_in,
                              void* d_out, int out_size, void* d_ws, size_t ws_size,
                              hipStream_t stream) {
    const float* q               = (const float*)d_in[0];
    const int*   predictions     = (const int*)d_in[3];
    const float* pos_queue       = (const float*)d_in[4];
    const float* neg_queue       = (const float*)d_in[5];
    const int*   queue_labels    = (const int*)d_in[6];
    const int*   queue_preds     = (const int*)d_in[7];

    const int B = in_sizes[3];               // 1024
    const int K = in_sizes[6];               // 131072
    const int D = in_sizes[0] / B;           // 128

    char* ws = (char*)d_ws;
    float*         sums   = (float*)(ws + OFF_SUMS);
    int*           counts = (int*)(ws + OFF_CNT);
    unsigned char* code   = (unsigned char*)(ws + OFF_CODE);
    float*         Mat    = (float*)(ws + OFF_MAT);
    float*         qn     = (float*)(ws + OFF_QN);
    float*         simout = (float*)(ws + OFF_SIM);

    // 1) zero accumulators
    zero_kernel<<<4, 1024, 0, stream>>>(sums, counts, 2 * NCLASS * D);
    // 2) class/correct codes + counts
    prep_kernel<<<(K + 255) / 256, 256, 0, stream>>>(queue_preds, queue_labels, code, counts, K);
    // 3) masked per-class column sums over both queues (memory-bound pass: 128 MB)
    colsum_kernel<<<dim3(D, 16, 2), 256, 0, stream>>>(pos_queue, neg_queue, code, sums, K, D);
    // 4) normalize q rows
    qnorm_kernel<<<B, D, 0, stream>>>(q, qn, D);
    // 5) assemble 128x32 RHS matrix
    matbuild_kernel<<<D, 32, 0, stream>>>(sums, Mat, D);
    // 6) WMMA GEMM: 128 waves (64 row tiles x 2 col halves), 8 waves/block
    wmma_gemm_kernel<<<(B / 16) * 2 / 8, 256, 0, stream>>>(qn, Mat, simout);
    // 7) finalize loss + accuracy
    finalize_kernel<<<1, 1024, 0, stream>>>(simout, counts, predictions, (float*)d_out, B);
}